// DifferentiableDAG_46986942218932
// MI455X (gfx1250) — compile-verified
//
#include <hip/hip_runtime.h>
#include <math.h>

// DifferentiableDAG: B=32, T=8192, D=8 steps, N=9 nodes, 5 ops.
// Memory-bound streaming kernel (~212 MB in / 1 MB out -> ~9us @ 23.3 TB/s).
// CDNA5 path used: global_load_async_to_lds_b128 tile staging + s_wait_asynccnt.

#define BLOCK 128   // threads per workgroup = 4 waves (wave32)
#define NN 9
#define DD 8
// per-thread contiguous chunks: p1 = 72 floats, p2 = 72 floats, pop = 40 floats
#define LDS_FLOATS (BLOCK * (72 + 72 + 40))   // 23552 floats = 94208 bytes

typedef int v4i __attribute__((ext_vector_type(4)));
typedef __attribute__((address_space(1))) v4i g_v4i;   // prints as "__device__" in HIP diags
typedef __attribute__((address_space(3))) v4i l_v4i;   // prints as "__shared__"

__device__ __forceinline__ void async_copy_b128(void* lds_dst, const void* gsrc) {
#if __has_builtin(__builtin_amdgcn_global_load_async_to_lds_b128)
  __builtin_amdgcn_global_load_async_to_lds_b128(
      (g_v4i*)(unsigned long long)(size_t)gsrc,
      (l_v4i*)(unsigned)(size_t)lds_dst,
      /*imm offset=*/0, /*cpol=*/0);
#else
  // VGLOBAL encoding, GV mode: vdst = LDS byte address, vaddr = 64-bit global addr.
  unsigned lds_addr = (unsigned)(size_t)lds_dst;  // low 32 bits of flat LDS addr = LDS offset
  asm volatile("global_load_async_to_lds_b128 %0, %1, off"
               :: "v"(lds_addr), "v"(gsrc) : "memory");
#endif
}

__device__ __forceinline__ void wait_async0() {
#if __has_builtin(__builtin_amdgcn_s_wait_asynccnt)
  __builtin_amdgcn_s_wait_asynccnt(0);
#else
  asm volatile("s_wait_asynccnt 0" ::: "memory");
#endif
}

__device__ __forceinline__ float tanh_fast(float x) {
#if __has_builtin(__builtin_amdgcn_tanhf)
  return __builtin_amdgcn_tanhf(x);   // v_tanh_f32 (gfx1250 TRANS)
#else
  return tanhf(x);
#endif
}

__device__ __forceinline__ float clip_log(float x) {
  return tanh_fast(x * (1.0f / 15.0f)) * 15.0f;
}

__global__ __launch_bounds__(BLOCK)
void dag_kernel(const float* __restrict__ isgn, const float* __restrict__ ilog,
                const float* __restrict__ p1g,  const float* __restrict__ p2g,
                const float* __restrict__ ppg,  float* __restrict__ out) {
  __shared__ float sm[LDS_FLOATS];
  float* sm1 = sm;                 // BLOCK * 72 : operand1 probs tile (flat copy of global)
  float* sm2 = sm + BLOCK * 72;    // BLOCK * 72 : operand2 probs tile
  float* smp = sm + BLOCK * 144;   // BLOCK * 40 : operation probs tile

  const int tid = threadIdx.x;
  const long long base = (long long)blockIdx.x * BLOCK;  // first flat (b,t) of tile
  const long long idx  = base + tid;

  // ---- Stage the tile's probability data into LDS (async DMA, 16B/lane, coalesced) ----
  {
    const float* g1 = p1g + base * 72;   // tile region: BLOCK*288 B contiguous, 16B aligned
    const float* g2 = p2g + base * 72;
    const float* gp = ppg + base * 40;   // tile region: BLOCK*160 B contiguous, 16B aligned
#pragma unroll
    for (int k = 0; k < 18; ++k) {       // BLOCK*18 b128 transfers for p1
      int u = tid + k * BLOCK;
      async_copy_b128(sm1 + u * 4, g1 + u * 4);
    }
#pragma unroll
    for (int k = 0; k < 18; ++k) {
      int u = tid + k * BLOCK;
      async_copy_b128(sm2 + u * 4, g2 + u * 4);
    }
#pragma unroll
    for (int k = 0; k < 10; ++k) {       // BLOCK*10 b128 transfers for op probs
      int u = tid + k * BLOCK;
      async_copy_b128(smp + u * 4, gp + u * 4);
    }
  }

  // ---- Overlap the DMA with the per-thread node initialization (9 contiguous floats) ----
  float s[NN], l[NN];
  {
    const float* gs = isgn + idx * NN;
    const float* gl = ilog + idx * NN;
#pragma unroll
    for (int j = 0; j < NN; ++j) { s[j] = gs[j]; l[j] = gl[j]; }
  }

  wait_async0();      // my wave's async writes to LDS complete
  __syncthreads();    // all waves' tiles visible

  const float* P1 = sm1 + tid * 72;
  const float* P2 = sm2 + tid * 72;
  const float* PP = smp + tid * 40;

  float ss = l[0] * l[0];  // running sum of squares of committed log nodes
#pragma unroll
  for (int st = 0; st < DD; ++st) {
    // four 9-length dots (per-element weights -> scalar FMAs, not a matmul)
    float s1 = 0.f, l1 = 0.f, s2 = 0.f, l2 = 0.f;
#pragma unroll
    for (int j = 0; j < NN; ++j) {
      float a = P1[st * 9 + j], b = P2[st * 9 + j];
      s1 = fmaf(a, s[j], s1);  l1 = fmaf(a, l[j], l1);
      s2 = fmaf(b, s[j], s2);  l2 = fmaf(b, l[j], l2);
    }

    // shared log-domain pieces for add/sub (ly identical for both -> CSE by hand)
    float mx = fmaxf(l1, l2);
    float mn = fminf(l1, l2);
    float l_same = clip_log(mx + log1pf(__expf(mn - mx)));   // clip(logaddexp(l1,l2))
    bool  bigger = (l1 >= l2);
    float big_l   = bigger ? l1 : l2;
    float small_l = bigger ? l2 : l1;
    float delta = fminf(fmaxf(small_l - big_l, -15.0f), -0.001f);
    float ldiff = log1pf(-__expf(delta));
    bool  zr = (small_l == big_l);
    float l_opp = zr ? 0.0f : (big_l + ldiff);

    bool  zx    = (s1 == 0.0f);
    float ssame = (s1 > 0.0f) ? 1.0f : -1.0f;  // sign(s1), only used when s1 != 0

    // ADD: y = (s2, l2)
    float sa, la;
    {
      float sy = s2;
      bool  zy   = (sy == 0.0f);
      bool  same = (s1 * sy > 0.0f);
      float new_s = zr ? 0.0f : (bigger ? s1 : sy);
      float s_o = zx ? (zy ? 0.0f : sy) : (zy ? s1 : (same ? ssame : new_s));
      float l_o = zx ? (zy ? 0.0f : l2) : (zy ? l1 : (same ? l_same : l_opp));
      sa = s_o; la = clip_log(l_o);
    }
    // SUB: y = (-s2, l2)
    float sb, lb;
    {
      float sy = -s2;
      bool  zy   = (sy == 0.0f);
      bool  same = (s1 * sy > 0.0f);
      float new_s = zr ? 0.0f : (bigger ? s1 : sy);
      float s_o = zx ? (zy ? 0.0f : sy) : (zy ? s1 : (same ? ssame : new_s));
      float l_o = zx ? (zy ? 0.0f : l2) : (zy ? l1 : (same ? l_same : l_opp));
      sb = s_o; lb = clip_log(l_o);
    }
    float s12  = s1 * s2;              // mul and div share the sign product
    float lmul = clip_log(l1 + l2);
    float ldiv = clip_log(l1 - l2);

    float pa = PP[st * 5 + 0], pu = PP[st * 5 + 1], pm = PP[st * 5 + 2],
          pd = PP[st * 5 + 3], pi = PP[st * 5 + 4];
    float smix = pa * sa + pu * sb + (pm + pd) * s12 + pi * s1;
    float lmix = pa * la + pu * lb + pm * lmul + pd * ldiv + pi * l1;
    lmix = clip_log(lmix);

    // RMS renorm over logs[0..st] ++ lmix  (st+2 values), scale applied to lmix only
    const float inv  = 1.0f / (float)(st + 2);   // compile-time constant per unrolled step
    float mean  = (ss + lmix * lmix) * inv;
    float rms   = sqrtf(mean + 1e-6f);
    float scale = fminf(15.0f / rms, 1.0f);
    lmix *= scale;

    s[st + 1] = smix;
    l[st + 1] = lmix;
    ss = fmaf(lmix, lmix, ss);
  }

  out[idx] = s[DD] * __expf(l[DD]);
}

extern "C" void kernel_launch(void* const* d_in, const int* in_sizes, int n_in,
                              void* d_out, int out_size, void* d_ws, size_t ws_size,
                              hipStream_t stream) {
  (void)n_in; (void)d_ws; (void)ws_size; (void)out_size;
  const float* isgn = (const float*)d_in[0];   // (B,T,9)
  const float* ilog = (const float*)d_in[1];   // (B,T,9)
  const float* p1   = (const float*)d_in[2];   // (B,T,8,9)
  const float* p2   = (const float*)d_in[3];   // (B,T,8,9)
  const float* pp   = (const float*)d_in[4];   // (B,T,8,5)
  float* out = (float*)d_out;                  // (B,T)

  const int BT = in_sizes[0] / NN;             // 32*8192 = 262144; divisible by BLOCK
  dim3 grid(BT / BLOCK), block(BLOCK);
  dag_kernel<<<grid, block, 0, stream>>>(isgn, ilog, p1, p2, pp, out);
}